// PostProcessor_46626164965464
// MI455X (gfx1250) — compile-verified
//
#include <hip/hip_runtime.h>
#include <hip/hip_bf16.h>

// HOI post-processor for MI455X (gfx1250, wave32).
// out[h,o,a] = hum[h]*obj[o]*logit[h,a]*exp(-(|enc(h,o)|^2+|mu(h,a)|^2-2 enc.mu)/(2*std^2))
// N=1500, A=64 actions, 81 detection classes, std=0.3, person class = 2.

typedef __attribute__((ext_vector_type(2))) float v2f;
typedef __attribute__((ext_vector_type(4))) float v4f;
typedef __attribute__((ext_vector_type(8))) float v8f;

#define NUM_CLASSES 81
#define NUM_ACTIONS 64
#define PERSON_IDX 2

// ---------------------------------------------------------------------------
// Kernel 1: per-proposal preprocessing (tiny: N threads).
// W layout (floats): [0]=cx [N]=cy [2N]=1/w [3N]=1/h [4N]=ln w [5N]=ln h
//                    [6N]=humaness [7N]=objectness
// ---------------------------------------------------------------------------
__global__ void hoi_pre(const float* __restrict__ scores,
                        const float* __restrict__ bbox,
                        float* __restrict__ W, int N) {
    int i = blockIdx.x * blockDim.x + threadIdx.x;
    if (i >= N) return;

    // argmax over 81 classes (first occurrence on ties, like jnp.argmax)
    const float* srow = scores + (size_t)i * NUM_CLASSES;
    float best = srow[0];
    int bidx = 0;
    for (int j = 1; j < NUM_CLASSES; ++j) {
        float v = srow[j];
        if (v > best) { best = v; bidx = j; }
    }

    float x0 = bbox[i * 4 + 0], y0 = bbox[i * 4 + 1];
    float x1 = bbox[i * 4 + 2], y1 = bbox[i * 4 + 3];
    float w = x1 - x0, h = y1 - y0;

    const float LN2 = 0.6931471805599453f;
    W[0 * N + i] = x0 + 0.5f * w;                        // cx
    W[1 * N + i] = y0 + 0.5f * h;                        // cy
    W[2 * N + i] = 1.0f / w;                             // inv w
    W[3 * N + i] = 1.0f / h;                             // inv h
    W[4 * N + i] = __builtin_amdgcn_logf(w) * LN2;       // ln w (v_log_f32 is log2)
    W[5 * N + i] = __builtin_amdgcn_logf(h) * LN2;       // ln h
    W[6 * N + i] = (bidx == PERSON_IDX) ? best : 0.0f;   // humaness
    W[7 * N + i] = (bidx == PERSON_IDX) ? 0.0f : best;   // objectness
}

// ---------------------------------------------------------------------------
// Kernel 2: one wave32 per (h, 16-wide o-tile).
//  A frag (enc, 16x4 f32): lane%16 = o, VGPR/lane-half = component per ISA layout:
//    v0 = enc[o, lane<16 ? 0 : 2], v1 = enc[o, lane<16 ? 1 : 3]
//  B frag (mu^T, 4x16 f32): lane%16 = a, v0 = mu[a, lane<16 ? 0 : 2], v1 = mu[a, .. 1:3]
//  D (16x16 f32): VGPR r, lane L -> o = obase + r + 8*(L/16), a = abase + L%16
// ---------------------------------------------------------------------------
__global__ __launch_bounds__(256) void hoi_main(
    const float* __restrict__ action_logits,  // [N,64]
    const float* __restrict__ target_mean,    // [N,64,4]
    const float* __restrict__ W,              // preprocessed arrays
    float* __restrict__ out,                  // [N,N,64]
    int N, int OT) {
    const int lane = threadIdx.x & 31;
    const int wave = blockIdx.x * (blockDim.x >> 5) + (threadIdx.x >> 5);
    if (wave >= N * OT) return;               // wave-uniform guard (EXEC stays all-1)

    const int h      = wave / OT;
    const int o_base = (wave - h * OT) << 4;

    const float* cx   = W + 0 * N;
    const float* cy   = W + 1 * N;
    const float* invw = W + 2 * N;
    const float* invh = W + 3 * N;
    const float* lw   = W + 4 * N;
    const float* lh   = W + 5 * N;
    const float* hum  = W + 6 * N;
    const float* obj  = W + 7 * N;

    const int half = lane >> 4;   // 0: components {0,1}, 1: components {2,3}
    const int ol   = lane & 15;
    int o = o_base + ol;
    if (o >= N) o = N - 1;        // clamp tail reads; stores are masked below

    // per-h scalars (uniform across wave -> broadcast loads)
    const float cxh = cx[h], cyh = cy[h], iwh = invw[h], ihh = invh[h];
    const float lwh = lw[h], lhh = lh[h], H = hum[h];

    // enc(h, o, :) — every lane computes all 4 components (cheap, gives e2 free)
    const float c0 = (cx[o] - cxh) * iwh;
    const float c1 = (cy[o] - cyh) * ihh;
    const float c2 = lw[o] - lwh;
    const float c3 = lh[o] - lhh;
    const float e2v  = c0 * c0 + c1 * c1 + c2 * c2 + c3 * c3;  // |enc|^2 for o=o_base+ol
    const float objv = obj[o];

    v2f afrag;
    afrag.x = half ? c2 : c0;
    afrag.y = half ? c3 : c1;

    // B fragments + |mu|^2 + logits for the 4 a-tiles (one b128 load per lane each)
    v2f   bfrag[4];
    float m2v[4], alv[4];
#pragma unroll
    for (int t = 0; t < 4; ++t) {
        const int a = t * 16 + ol;
        const v4f mu = *(const v4f*)(target_mean + ((size_t)h * NUM_ACTIONS + a) * 4);
        bfrag[t].x = half ? mu.z : mu.x;
        bfrag[t].y = half ? mu.w : mu.y;
        m2v[t] = mu.x * mu.x + mu.y * mu.y + mu.z * mu.z + mu.w * mu.w;
        alv[t] = action_logits[(size_t)h * NUM_ACTIONS + a];
    }

    // cross[h, o_tile, a_tile] via V_WMMA_F32_16X16X4_F32 (exact shape match, K=4)
    v8f acc[4];
#pragma unroll
    for (int t = 0; t < 4; ++t) {
        v8f c = {};
        acc[t] = __builtin_amdgcn_wmma_f32_16x16x4_f32(
            /*neg_a=*/false, afrag, /*neg_b=*/false, bfrag[t],
            /*c_mod=*/(short)0, c, /*reuse_a=*/false, /*reuse_b=*/false);
    }

    // epilogue: compat * scores, non-temporal coalesced b32 stores
    const float K2 = (1.0f / (2.0f * 0.3f * 0.3f)) * 1.4426950408889634f; // log2e/(2s^2)
#pragma unroll
    for (int r = 0; r < 8; ++r) {
        const int src = r + (half << 3);            // lane holding this row's o data
        const float e2r  = __shfl(e2v, src, 32);    // ds_bpermute broadcast
        const float objr = __shfl(objv, src, 32);
        const int orow = o_base + r + (half << 3);
        const bool valid = orow < N;
        const float s = H * objr;
        float* rowp = out + ((size_t)h * N + orow) * NUM_ACTIONS + ol;
#pragma unroll
        for (int t = 0; t < 4; ++t) {
            const float cr  = acc[t][r];
            const float arg = (2.0f * cr - e2r - m2v[t]) * K2;
            const float val = s * alv[t] * __builtin_amdgcn_exp2f(arg);
            if (valid) __builtin_nontemporal_store(val, rowp + t * 16);
        }
    }
}

// ---------------------------------------------------------------------------
extern "C" void kernel_launch(void* const* d_in, const int* in_sizes, int n_in,
                              void* d_out, int out_size, void* d_ws, size_t ws_size,
                              hipStream_t stream) {
    const float* action_logits = (const float*)d_in[0];  // [N,64]
    const float* target_mean   = (const float*)d_in[1];  // [N,64,4]
    const float* bbox          = (const float*)d_in[2];  // [N,4]
    const float* scores        = (const float*)d_in[3];  // [N,81]
    float* out = (float*)d_out;
    float* W   = (float*)d_ws;                           // 8*N floats

    const int N = in_sizes[2] / 4;                       // 1500
    hoi_pre<<<(N + 255) / 256, 256, 0, stream>>>(scores, bbox, W, N);

    const int OT = (N + 15) / 16;                        // 94 o-tiles
    const int total_waves = N * OT;                      // 141,000
    const int wpb = 256 / 32;                            // 8 waves per block
    const int grid = (total_waves + wpb - 1) / wpb;      // 17,625 (exact)
    hoi_main<<<grid, 256, 0, stream>>>(action_logits, target_mean, W, out, N, OT);
}